// HoughVoting_35141422416214
// MI455X (gfx1250) — compile-verified
//
#include <hip/hip_runtime.h>
#include <math.h>

typedef __attribute__((ext_vector_type(2))) float v2f;
typedef __attribute__((ext_vector_type(4))) float v4f;
typedef __attribute__((ext_vector_type(8))) float v8f;

#define HWT     307200      // 480*640
#define WIDTH   640
#define SKIPPX  10
#define NP      30720       // HWT / SKIPPX
#define NTILE   1920        // NP / 16
#define NCAND   768         // 24 * 32 candidate centers (stride 20)
#define NCBLK   48          // NCAND / 16
#define NCLS    22
#define VOTES_N (NCLS * NCAND)        // 16896
#define WS_FLOATS (2 * VOTES_N + 32)  // votes | zsum | count
#define ATAB_V4  (NCBLK * 32)         // per-(blk,lane) A-matrix cache (float4)
#define SMEM_FLOATS (WS_FLOATS + 4 * ATAB_V4)
#define NWG 120                       // 120 WG * 8 waves * 2 tiles = 1920

__global__ void hv_init(float* ws) {
    int i = blockIdx.x * blockDim.x + threadIdx.x;
    if (i < WS_FLOATS) ws[i] = 0.0f;
}

__global__ __launch_bounds__(256) void hv_vote(const int* __restrict__ label,
                                               const float* __restrict__ vertex,
                                               float* __restrict__ ws) {
    extern __shared__ float smem[];
    float* votes = smem;                 // [NCLS*NCAND]
    float* zsum  = smem + VOTES_N;       // [NCLS*NCAND]
    float* cnt   = smem + 2 * VOTES_N;   // [32]
    v4f*   atab  = (v4f*)(smem + WS_FLOATS);  // [NCBLK*32] (a1x,a1y,a2x,a2y)

    for (int i = threadIdx.x; i < WS_FLOATS; i += blockDim.x) smem[i] = 0.0f;

    // Precompute candidate A-matrix features per (blk, lane-position).
    // A layout (16x4 f32): lanes0-15 hold K0/K1 (M=lane), lanes16-31 hold K2/K3 (M=lane-16).
    for (int idx = threadIdx.x; idx < ATAB_V4; idx += blockDim.x) {
        int lp   = idx & 31;
        int blk  = idx >> 5;
        int lcol = lp & 15;
        int lhal = lp >> 4;
        int cand0 = blk * 16 + lcol;
        float cxf = (float)((cand0 & 31) * 20);
        float cyf = (float)((cand0 >> 5) * 20);
        v4f a;
        if (lhal == 0) {
            a.x = cxf;                    // dot_un A: (cx, cy | 1, 0)
            a.y = cyf;
            a.z = cxf * cxf + cyf * cyf;  // dd2 A: (cx^2+cy^2, -2cx | -2cy, 1)
            a.w = -2.0f * cxf;
        } else {
            a.x = 1.0f;
            a.y = 0.0f;
            a.z = -2.0f * cyf;
            a.w = 1.0f;
        }
        atab[idx] = a;
    }
    __syncthreads();

    const int lane   = threadIdx.x & 31;
    const int wave   = threadIdx.x >> 5;
    const int gwave  = blockIdx.x * (blockDim.x >> 5) + wave;
    const int nwaves = gridDim.x * (blockDim.x >> 5);
    const int col    = lane & 15;   // N column (pixel)
    const int half   = lane >> 4;   // selects K{0,1} vs K{2,3} in A/B layouts

    for (int tile = gwave; tile < NTILE; tile += nwaves) {
        // ---- per-lane pixel features (column N = col) ----
        int   j  = tile * 16 + col;
        int   sv = j * SKIPPX;
        float px = (float)(sv % WIDTH);
        float py = (float)(sv / WIDTH);
        int   c  = label[sv];
        float u = 0.0f, v = 0.0f, z = 0.0f;
        int   m = 0;
        if (c > 0) {
            const float* vb = vertex + (size_t)(3 * c) * HWT + sv;
            u = vb[0]; v = vb[HWT]; z = vb[2 * (size_t)HWT];
            float n = sqrtf(u * u + v * v) + 1e-12f;
            u /= n; v /= n;
            m = 1;
        }
        float w  = -(px * u + py * v);
        float pp = px * px + py * py;

        // B matrices (4x16): lanes0-15 hold K0/K1, lanes16-31 hold K2/K3
        float b1x = (half == 0) ? u    : w;     // (u, v, w, 0)
        float b1y = (half == 0) ? v    : 0.0f;
        float b2x = (half == 0) ? 1.0f : py;    // (1, px, py, px^2+py^2)
        float b2y = (half == 0) ? px   : pp;
        v2f b1 = {b1x, b1y};
        v2f b2 = {b2x, b2y};

        if (half == 0 && c > 0) atomicAdd(&cnt[c], 1.0f);  // pixel counted once

        const int cbase = c * NCAND;
        for (int blk = 0; blk < NCBLK; ++blk) {
            v4f a = atab[blk * 32 + lane];   // ds_load_b128
            v2f a1 = {a.x, a.y};
            v2f a2 = {a.z, a.w};

            v8f zero = {};
            // dot_un = (cx-px)*u + (cy-py)*v ;  dd2 = (cx-px)^2 + (cy-py)^2
            v8f dotm = __builtin_amdgcn_wmma_f32_16x16x4_f32(
                false, a1, false, b1, (short)0, zero, false, false);
            v8f dd2m = __builtin_amdgcn_wmma_f32_16x16x4_f32(
                false, a2, false, b2, (short)0, zero, false, false);

            if (m) {
                #pragma unroll
                for (int e = 0; e < 8; ++e) {
                    float du = dotm[e];
                    float d2 = dd2m[e];
                    // dot/dd > 0.9  &&  dd > 0   (squared form)
                    if (d2 > 0.0f && du > 0.0f && du * du > 0.81f * d2) {
                        int candM = blk * 16 + e + half * 8;  // D row layout
                        atomicAdd(&votes[cbase + candM], 1.0f);
                        atomicAdd(&zsum [cbase + candM], z);
                    }
                }
            }
        }
    }
    __syncthreads();
    // flush workgroup partials to global accumulators (sparse: skip zeros)
    for (int i = threadIdx.x; i < WS_FLOATS; i += blockDim.x) {
        float t = smem[i];
        if (t != 0.0f) atomicAdd(&ws[i], t);
    }
}

__global__ void hv_final(const float* __restrict__ ws,
                         const float* __restrict__ meta,
                         const float* __restrict__ extents,
                         float* __restrict__ out) {
    const float* votes = ws;
    const float* zsum  = ws + VOTES_N;
    const float* cnt   = ws + 2 * VOTES_N;
    int lane = threadIdx.x & 31;
    int wv   = threadIdx.x >> 5;
    if (wv >= NCLS - 1) return;
    int c = wv + 1;

    // per-lane scan (stride 32) keeping lowest index on ties
    float bestv = -1.0f; int besti = 0;
    for (int i = lane; i < NCAND; i += 32) {
        float vv = votes[c * NCAND + i];
        if (vv > bestv) { bestv = vv; besti = i; }
    }
    // wave32 reduction, first-max tie-break
    for (int off = 16; off > 0; off >>= 1) {
        float ov = __shfl_xor(bestv, off, 32);
        int   oi = __shfl_xor(besti, off, 32);
        if (ov > bestv || (ov == bestv && oi < besti)) { bestv = ov; besti = oi; }
    }
    if (lane == 0) {
        float maxv  = bestv;
        float bx    = (float)((besti & 31) * 20);
        float by    = (float)((besti >> 5) * 20);
        float depth = expf(zsum[c * NCAND + besti] / fmaxf(maxv, 1.0f));
        float fx = meta[0], px0 = meta[2], fy = meta[4], py0 = meta[5];
        float vf = (cnt[c] > 100.0f && maxv > 0.0f) ? 1.0f : 0.0f;
        float tx = (bx - px0) / fx * depth;
        float ty = (by - py0) / fy * depth;
        float e0 = extents[c * 3 + 0], e1 = extents[c * 3 + 1], e2 = extents[c * 3 + 2];
        float diam = sqrtf(e0 * e0 + e1 * e1 + e2 * e2);
        float bw = diam * fx / depth;
        float bh = diam * fy / depth;
        float* box = out + (c - 1) * 7;
        box[0] = 0.0f;
        box[1] = (float)c * vf;
        box[2] = (bx - bw * 0.5f) * vf;
        box[3] = (by - bh * 0.5f) * vf;
        box[4] = (bx + bw * 0.5f) * vf;
        box[5] = (by + bh * 0.5f) * vf;
        box[6] = maxv * vf;
        float* pose = out + 147 + (c - 1) * 7;
        pose[0] = vf; pose[1] = 0.0f; pose[2] = 0.0f; pose[3] = 0.0f;
        pose[4] = tx * vf; pose[5] = ty * vf; pose[6] = depth * vf;
    }
}

extern "C" void kernel_launch(void* const* d_in, const int* in_sizes, int n_in,
                              void* d_out, int out_size, void* d_ws, size_t ws_size,
                              hipStream_t stream) {
    const int*   label   = (const int*)d_in[0];
    const float* vertex  = (const float*)d_in[1];
    const float* meta    = (const float*)d_in[2];
    const float* extents = (const float*)d_in[3];
    float* out = (float*)d_out;
    float* ws  = (float*)d_ws;

    hv_init<<<(WS_FLOATS + 255) / 256, 256, 0, stream>>>(ws);
    size_t lds_bytes = (size_t)SMEM_FLOATS * sizeof(float);  // ~159 KB, fits 320 KB WGP LDS
    hv_vote<<<NWG, 256, lds_bytes, stream>>>(label, vertex, ws);
    hv_final<<<1, (NCLS - 1) * 32, 0, stream>>>(ws, meta, extents, out);
}